// FPN2MLPFeatureExtractorNeighbor_17678085391109
// MI455X (gfx1250) — compile-verified
//
#include <hip/hip_runtime.h>
#include <hip/hip_bf16.h>

typedef float v2f __attribute__((ext_vector_type(2)));
typedef float v8f __attribute__((ext_vector_type(8)));

#define N_BOX 512
#define CH    256
#define RES   7
#define HW    49      // 7*7
#define S_GRID 14     // RES*RATIO
#define FLAT  12544   // CH*HW
#define NLIC  128
#define NLG   4
#define NLCG  32
#define REP   1024

// ---------------------------------------------------------------------------
// K1: ROI-align + FPN level select.  One block per box, thread = channel.
// Bilinear coordinates are separable (x depends on j, y on i) -> precompute
// 14 per-axis coord sets in LDS, then each channel does 49 cells * 4 taps.
// ---------------------------------------------------------------------------
__global__ void roi_pool_kernel(const float* __restrict__ f0, const float* __restrict__ f1,
                                const float* __restrict__ f2, const float* __restrict__ f3,
                                const float* __restrict__ boxes, const int* __restrict__ bidx,
                                float* __restrict__ xout) {
  const int n = blockIdx.x;
  const int tid = threadIdx.x;

  const float x1 = boxes[n*4+0], y1 = boxes[n*4+1];
  const float x2 = boxes[n*4+2], y2 = boxes[n*4+3];
  const float area = fmaxf(x2-x1, 0.f) * fmaxf(y2-y1, 0.f);
  float lv = floorf(4.0f + log2f(sqrtf(area) / 224.0f + 1e-6f));
  lv = fminf(fmaxf(lv, 2.0f), 5.0f);
  const int lvl = (int)lv - 2;

  const float scales[4] = {0.25f, 0.125f, 0.0625f, 0.03125f};
  const int   Hs[4] = {200, 100, 50, 25};
  const int   Ws[4] = {304, 152, 76, 38};
  const float* feats[4] = {f0, f1, f2, f3};
  const float sc = scales[lvl];
  const int H = Hs[lvl], W = Ws[lvl];
  const float* feat = feats[lvl];

  const float bx1 = x1*sc, by1 = y1*sc, bx2 = x2*sc, by2 = y2*sc;
  const float cx = (bx1+bx2)*0.5f, cy = (by1+by2)*0.5f;
  const float bw = fmaxf(bx2-bx1, 1.0f) * 2.0f;
  const float bh = fmaxf(by2-by1, 1.0f) * 2.0f;
  const float x0b = cx - bw*0.5f, y0b = cy - bh*0.5f;

  __shared__ int   sx0i[S_GRID], sx1i[S_GRID], sy0i[S_GRID], sy1i[S_GRID];
  __shared__ float slx[S_GRID], sly[S_GRID];
  if (tid < S_GRID) {
    const float off = ((float)tid + 0.5f) / (float)S_GRID;
    float xx = x0b + off*bw;
    float x0 = fminf(fmaxf(floorf(xx), 0.f), (float)(W-1));
    slx[tid] = fminf(fmaxf(xx - x0, 0.f), 1.f);
    int x0i = (int)x0;
    sx0i[tid] = x0i;
    sx1i[tid] = min(x0i + 1, W - 1);
    float yy = y0b + off*bh;
    float y0 = fminf(fmaxf(floorf(yy), 0.f), (float)(H-1));
    sly[tid] = fminf(fmaxf(yy - y0, 0.f), 1.f);
    int y0i = (int)y0;
    sy0i[tid] = y0i;
    sy1i[tid] = min(y0i + 1, H - 1);
  }
  __syncthreads();

  const int b = bidx[n];
  const float* fc = feat + ((size_t)b*CH + tid) * (size_t)(H*W);
  float* xo = xout + (size_t)n*FLAT + (size_t)tid*HW;

  for (int cell = 0; cell < HW; ++cell) {
    const int py = cell / RES, px = cell % RES;
    float acc = 0.f;
    #pragma unroll
    for (int ry = 0; ry < 2; ++ry) {
      const int i = py*2 + ry;
      const int y0 = sy0i[i], y1v = sy1i[i];
      const float ly = sly[i];
      #pragma unroll
      for (int rx = 0; rx < 2; ++rx) {
        const int j = px*2 + rx;
        const int x0 = sx0i[j], x1v = sx1i[j];
        const float lx = slx[j];
        const float v00 = fc[y0*W + x0],  v01 = fc[y0*W + x1v];
        const float v10 = fc[y1v*W + x0], v11 = fc[y1v*W + x1v];
        acc += v00*(1.f-ly)*(1.f-lx) + v01*(1.f-ly)*lx
             + v10*ly*(1.f-lx)       + v11*ly*lx;
      }
    }
    xo[cell] = acc * 0.25f;
  }
}

// ---------------------------------------------------------------------------
// K2: non-local block, one block (256 threads) per box.  Streams one group at
// a time through LDS (t/p/g 32x49 each, attn 49x49, y 32x49 = 34.7 KB), keeps
// the z residual accumulator in registers (thread = output channel, 49 regs),
// writes z in place over the pooled buffer.
// ---------------------------------------------------------------------------
__global__ void nonlocal_kernel(float* __restrict__ xbuf,
                                const float* __restrict__ tw, const float* __restrict__ tb,
                                const float* __restrict__ pw, const float* __restrict__ pb,
                                const float* __restrict__ gw, const float* __restrict__ gb,
                                const float* __restrict__ ww, const float* __restrict__ wb) {
  const int n = blockIdx.x;
  const int tid = threadIdx.x;
  float* xb = xbuf + (size_t)n*FLAT;

  __shared__ float st[NLCG*HW], sp[NLCG*HW], sg[NLCG*HW], sy[NLCG*HW];
  __shared__ float sa[HW*HW];

  // z = w_b + x (residual) accumulator; thread tid owns output channel tid
  float z[HW];
  {
    const float* xr = xb + tid*HW;
    const float bv = wb[tid];
    #pragma unroll
    for (int s = 0; s < HW; ++s) z[s] = bv + xr[s];
  }

  for (int g = 0; g < NLG; ++g) {
    // theta / phi / g embeddings for this group: 3 * 32x49 outputs, K=256
    for (int idx = tid; idx < 3*NLCG*HW; idx += 256) {
      const int m = idx / (NLCG*HW);
      const int r = idx % (NLCG*HW);
      const int i = r / HW, s = r % HW;
      const float* Wm = (m == 0) ? tw : ((m == 1) ? pw : gw);
      const float* Bm = (m == 0) ? tb : ((m == 1) ? pb : gb);
      const int row = g*NLCG + i;
      const float* wrow = Wm + (size_t)row*CH;
      float acc = Bm[row];
      for (int c = 0; c < CH; ++c) acc += wrow[c] * xb[c*HW + s];
      float* dst = (m == 0) ? st : ((m == 1) ? sp : sg);
      dst[r] = acc;
    }
    __syncthreads();

    // attention logits: sa[s][t] = sum_c t[c][s] * p[c][t]
    for (int idx = tid; idx < HW*HW; idx += 256) {
      const int s = idx / HW, t = idx % HW;
      float acc = 0.f;
      #pragma unroll 8
      for (int c = 0; c < NLCG; ++c) acc += st[c*HW + s] * sp[c*HW + t];
      sa[idx] = acc;
    }
    __syncthreads();

    // softmax over t for each s
    if (tid < HW) {
      float* row = sa + tid*HW;
      float mx = row[0];
      for (int t = 1; t < HW; ++t) mx = fmaxf(mx, row[t]);
      float sum = 0.f;
      for (int t = 0; t < HW; ++t) { const float e = __expf(row[t] - mx); row[t] = e; sum += e; }
      const float inv = 1.0f / sum;
      for (int t = 0; t < HW; ++t) row[t] *= inv;
    }
    __syncthreads();

    // y[i][s] = sum_t attn[s][t] * g[i][t]
    for (int idx = tid; idx < NLCG*HW; idx += 256) {
      const int i = idx / HW, s = idx % HW;
      float acc = 0.f;
      #pragma unroll 7
      for (int t = 0; t < HW; ++t) acc += sa[s*HW + t] * sg[i*HW + t];
      sy[idx] = acc;
    }
    __syncthreads();

    // z[co][s] += sum_c w_w[co][g*32+c] * y[c][s]
    {
      const float* wrow = ww + (size_t)tid*NLIC + g*NLCG;
      for (int c = 0; c < NLCG; ++c) {
        const float w = wrow[c];
        const float* yr = sy + c*HW;
        #pragma unroll
        for (int s = 0; s < HW; ++s) z[s] += w * yr[s];
      }
    }
    __syncthreads();
  }

  float* zo = xb + tid*HW;
  #pragma unroll
  for (int s = 0; s < HW; ++s) zo[s] = z[s];
}

// ---------------------------------------------------------------------------
// K3/K4: out = relu(X @ W^T + bias) via V_WMMA_F32_16X16X4_F32 (exact fp32).
// Block = 256 threads = 8 waves; each wave owns a 16x16 tile of a 32x64
// macro-tile.  A (16x4) layout: lane L -> row m0+L%16, K pair k+2*(L>>4);
// B (4x16) mirrors it from row-major W.  Both operands are contiguous
// float2 loads.  D: lane L, vgpr v -> row m0+8*(L>>4)+v, col n0+L%16.
// ---------------------------------------------------------------------------
__global__ void gemm_bias_relu_wmma(const float* __restrict__ X,
                                    const float* __restrict__ Wt,
                                    const float* __restrict__ bias,
                                    float* __restrict__ out,
                                    int N, int K) {
  const int tid  = threadIdx.x;
  const int lane = tid & 31;
  const int wv   = tid >> 5;                    // 0..7
  const int m0   = blockIdx.y * 32 + (wv >> 2) * 16;
  const int n0   = blockIdx.x * 64 + (wv & 3) * 16;
  const int r    = lane & 15;
  const int kh   = (lane >> 4) << 1;            // 0 or 2

  const float* ap = X  + (size_t)(m0 + r) * K + kh;
  const float* bp = Wt + (size_t)(n0 + r) * K + kh;

  v8f acc = {};
  #pragma unroll 4
  for (int k = 0; k < K; k += 4) {
    const v2f a = *(const v2f*)(ap + k);
    const v2f b = *(const v2f*)(bp + k);
    acc = __builtin_amdgcn_wmma_f32_16x16x4_f32(
        /*neg_a=*/false, a, /*neg_b=*/false, b,
        /*c_mod=*/(short)0, acc, /*reuse_a=*/false, /*reuse_b=*/false);
  }

  const int orow = m0 + ((lane >> 4) << 3);
  const int ocol = n0 + r;
  const float bv = bias[ocol];
  #pragma unroll
  for (int v = 0; v < 8; ++v) {
    const float val = acc[v] + bv;
    out[(size_t)(orow + v) * N + ocol] = fmaxf(val, 0.0f);
  }
}

// ---------------------------------------------------------------------------
extern "C" void kernel_launch(void* const* d_in, const int* in_sizes, int n_in,
                              void* d_out, int out_size, void* d_ws, size_t ws_size,
                              hipStream_t stream) {
  const float* f0      = (const float*)d_in[0];
  const float* f1      = (const float*)d_in[1];
  const float* f2      = (const float*)d_in[2];
  const float* f3      = (const float*)d_in[3];
  const float* boxes   = (const float*)d_in[4];
  const int*   bidx    = (const int*)d_in[5];
  const float* theta_w = (const float*)d_in[6];
  const float* theta_b = (const float*)d_in[7];
  const float* phi_w   = (const float*)d_in[8];
  const float* phi_b   = (const float*)d_in[9];
  const float* g_w     = (const float*)d_in[10];
  const float* g_b     = (const float*)d_in[11];
  const float* w_w     = (const float*)d_in[12];
  const float* w_b     = (const float*)d_in[13];
  const float* fc6_w   = (const float*)d_in[14];
  const float* fc6_b   = (const float*)d_in[15];
  const float* fc7_w   = (const float*)d_in[16];
  const float* fc7_b   = (const float*)d_in[17];

  float* xbuf = (float*)d_ws;                       // 512*12544 floats (pooled -> z, in place)
  float* h6   = xbuf + (size_t)N_BOX * FLAT;        // 512*1024 floats

  roi_pool_kernel<<<N_BOX, 256, 0, stream>>>(f0, f1, f2, f3, boxes, bidx, xbuf);

  nonlocal_kernel<<<N_BOX, 256, 0, stream>>>(xbuf, theta_w, theta_b, phi_w, phi_b,
                                             g_w, g_b, w_w, w_b);

  // fc6: (512 x 12544) @ (1024 x 12544)^T, ReLU
  gemm_bias_relu_wmma<<<dim3(REP/64, N_BOX/32), 256, 0, stream>>>(
      xbuf, fc6_w, fc6_b, h6, REP, FLAT);

  // fc7: (512 x 1024) @ (1024 x 1024)^T, ReLU -> d_out
  gemm_bias_relu_wmma<<<dim3(REP/64, N_BOX/32), 256, 0, stream>>>(
      h6, fc7_w, fc7_b, (float*)d_out, REP, REP);
}